// DeepStreamOutput_29119878267620
// MI455X (gfx1250) — compile-verified
//
#include <hip/hip_runtime.h>

#define NCLS 80
#define NA 25200
#define NBATCH 4
#define CONF 0.25f
#define IOU_THR 0.45f
#define MAX_DET 100
#define ATTN_RES 14
#define NUM_BASE 5
#define TOPK 1024
#define ROI_SCALE 0.25f
#define NBINS 2048
#define CAND_MAX 4096
#define FH 160
#define FW 160
#define OUT_ROW (6 + FH*FW)
#define NEG_BIG (-3.402823e38f)
#define ATTN_LEN (NUM_BASE * ATTN_RES * ATTN_RES)   // 980

typedef __attribute__((ext_vector_type(2))) float v2f;
typedef __attribute__((ext_vector_type(8))) float v8f;

// ---------------------------------------------------------------------------
// Kernel 1: box decode via V_WMMA_F32_16X16X4_F32.
// D(16x16) = A(16x4: 16 anchors x [cx,cy,w,h]) * B(4x16: conv matrix, cols 4..15 = 0)
// A layout (32-bit A 16x4): lanes 0-15 hold K=0,1 ; lanes 16-31 hold K=2,3.
// B layout (4x16): VGPR0 = rows K=0 (lanes0-15) / K=2 (lanes16-31), VGPR1 = K=1/K=3.
// D layout: VGPR v -> M=v (lanes0-15, N=lane) and M=v+8 (lanes16-31, N=lane-16).
// ---------------------------------------------------------------------------
__global__ __launch_bounds__(256) void k_boxes_wmma(const float* __restrict__ test,
                                                    float* __restrict__ boxes) {
  const int wave = threadIdx.x >> 5;
  const int lane = threadIdx.x & 31;
  const int t = blockIdx.x * 8 + wave;       // tile id, wave-uniform
  const int TILES = NBATCH * (NA / 16);      // 25200 % 16 == 0
  if (t >= TILES) return;                    // uniform per wave -> EXEC stays full
  const int b  = t / (NA / 16);
  const int a0 = (t % (NA / 16)) * 16;
  const int m  = lane & 15;
  const int hi = lane >> 4;
  const int kb = hi * 2;

  const float* p = test + ((size_t)b * NA + a0 + m) * 85 + kb;
  v2f afrag; afrag.x = p[0]; afrag.y = p[1];

  const float convM[4][4] = {{ 1.f , 0.f , 1.f , 0.f},
                             { 0.f , 1.f , 0.f , 1.f},
                             {-0.5f, 0.f , 0.5f, 0.f},
                             { 0.f ,-0.5f, 0.f , 0.5f}};
  const int n = m;  // column index for this lane (both halves)
  v2f bfrag;
  bfrag.x = (n < 4) ? convM[kb][n]     : 0.f;
  bfrag.y = (n < 4) ? convM[kb + 1][n] : 0.f;

  v8f c = {};
  v8f d = __builtin_amdgcn_wmma_f32_16x16x4_f32(false, afrag, false, bfrag,
                                                (short)0, c, false, false);
  if (n < 4) {
#pragma unroll
    for (int v = 0; v < 8; ++v) {
      const int mm = v + 8 * hi;
      boxes[((size_t)b * NA + a0 + mm) * 4 + n] = d[v];
    }
  }
}

// ---------------------------------------------------------------------------
// Kernel 2: per-batch score histogram over (CONF, 1].
// ---------------------------------------------------------------------------
__global__ __launch_bounds__(256) void k_hist(const float* __restrict__ test,
                                              int* __restrict__ hist) {
  __shared__ int lh[NBINS];
  const int b = blockIdx.y;
  const int a = blockIdx.x * blockDim.x + threadIdx.x;
  for (int i = threadIdx.x; i < NBINS; i += blockDim.x) lh[i] = 0;
  __syncthreads();
  if (a < NA) {
    const float* row = test + ((size_t)b * NA + a) * 85;
    const float obj = row[4];
    for (int c = 0; c < NCLS; ++c) {
      const float s = row[5 + c] * obj;
      if (s > CONF) {
        int bin = (int)((s - CONF) * ((float)NBINS / (1.0f - CONF)));
        bin = bin < 0 ? 0 : (bin > NBINS - 1 ? NBINS - 1 : bin);
        atomicAdd(&lh[bin], 1);
      }
    }
  }
  __syncthreads();
  for (int i = threadIdx.x; i < NBINS; i += blockDim.x)
    if (lh[i]) atomicAdd(&hist[b * NBINS + i], lh[i]);
}

// ---------------------------------------------------------------------------
// Kernel 3: suffix-scan histogram -> threshold bin containing the TOPK cutoff.
// ---------------------------------------------------------------------------
__global__ void k_thresh(const int* __restrict__ hist, int* __restrict__ thresh) {
  const int b = threadIdx.x;
  if (b >= NBATCH) return;
  int cum = 0, T = 0;
  for (int bin = NBINS - 1; bin >= 0; --bin) {
    cum += hist[b * NBINS + bin];
    if (cum >= TOPK) { T = bin; break; }
  }
  thresh[b] = T;
}

// ---------------------------------------------------------------------------
// Kernel 4: collect candidates whose bin >= threshold bin.
// ---------------------------------------------------------------------------
__global__ __launch_bounds__(256) void k_collect(const float* __restrict__ test,
                                                 const int* __restrict__ thresh,
                                                 int* __restrict__ count,
                                                 float* __restrict__ cand_s,
                                                 int* __restrict__ cand_i) {
  const int b = blockIdx.y;
  const int a = blockIdx.x * blockDim.x + threadIdx.x;
  if (a >= NA) return;
  const int T = thresh[b];
  const float* row = test + ((size_t)b * NA + a) * 85;
  const float obj = row[4];
  for (int c = 0; c < NCLS; ++c) {
    const float s = row[5 + c] * obj;
    if (s > CONF) {
      int bin = (int)((s - CONF) * ((float)NBINS / (1.0f - CONF)));
      bin = bin < 0 ? 0 : (bin > NBINS - 1 ? NBINS - 1 : bin);
      if (bin >= T) {
        const int pos = atomicAdd(&count[b], 1);
        if (pos < CAND_MAX) {
          cand_s[b * CAND_MAX + pos] = s;
          cand_i[b * CAND_MAX + pos] = a * NCLS + c;
        }
      }
    }
  }
}

// ---------------------------------------------------------------------------
// Kernel 5: per-batch bitonic sort (descending) of up to CAND_MAX candidates
// in LDS; keep the TOPK best. 32 KB LDS, easy on CDNA5's 320 KB/WGP.
// ---------------------------------------------------------------------------
__global__ __launch_bounds__(1024) void k_sortk(const float* __restrict__ cand_s,
                                                const int* __restrict__ cand_i,
                                                const int* __restrict__ count,
                                                float* __restrict__ top_s,
                                                int* __restrict__ top_i) {
  __shared__ float ss[CAND_MAX];
  __shared__ int   ii[CAND_MAX];
  const int b = blockIdx.x;
  const int t = threadIdx.x;
  int cnt = count[b]; if (cnt > CAND_MAX) cnt = CAND_MAX;
  for (int i = t; i < CAND_MAX; i += blockDim.x) {
    if (i < cnt) { ss[i] = cand_s[b * CAND_MAX + i]; ii[i] = cand_i[b * CAND_MAX + i]; }
    else         { ss[i] = NEG_BIG;                  ii[i] = 0; }
  }
  __syncthreads();
  for (int k = 2; k <= CAND_MAX; k <<= 1) {
    for (int j = k >> 1; j > 0; j >>= 1) {
      for (int i = t; i < CAND_MAX; i += blockDim.x) {
        const int ixj = i ^ j;
        if (ixj > i) {
          const bool up = (i & k) == 0;           // descending blocks
          const float si = ss[i], sx = ss[ixj];
          if (up ? (si < sx) : (si > sx)) {
            ss[i] = sx; ss[ixj] = si;
            const int tmp = ii[i]; ii[i] = ii[ixj]; ii[ixj] = tmp;
          }
        }
      }
      __syncthreads();
    }
  }
  if (t < TOPK) { top_s[b * TOPK + t] = ss[t]; top_i[b * TOPK + t] = ii[t]; }
}

// ---------------------------------------------------------------------------
// Kernel 6: greedy class-aware NMS, one workgroup per batch (1024 threads,
// LDS argmax reductions). Writes boxes/score/cls into d_out and didx to ws.
// ---------------------------------------------------------------------------
__global__ __launch_bounds__(1024) void k_nms(const float* __restrict__ top_s_g,
                                              const int* __restrict__ top_i_g,
                                              const float* __restrict__ boxes,
                                              float* __restrict__ out,
                                              int* __restrict__ didx_out) {
  __shared__ float ssh[TOPK];
  __shared__ int   cls[TOPK];
  __shared__ int   anc[TOPK];
  __shared__ float bx[TOPK][4];
  __shared__ float area[TOPK];
  __shared__ unsigned char valid[TOPK];
  __shared__ float red_s[TOPK];
  __shared__ int   red_i[TOPK];
  const int b = blockIdx.x;
  const int t = threadIdx.x;

  const float s = top_s_g[b * TOPK + t];
  const int idx = top_i_g[b * TOPK + t];
  const int an = idx / NCLS;
  ssh[t] = s; cls[t] = idx % NCLS; anc[t] = an;
  const float* bb = boxes + ((size_t)b * NA + an) * 4;
  const float x1 = bb[0], y1 = bb[1], x2 = bb[2], y2 = bb[3];
  bx[t][0] = x1; bx[t][1] = y1; bx[t][2] = x2; bx[t][3] = y2;
  float w = x2 - x1; if (w < 0.f) w = 0.f;
  float h = y2 - y1; if (h < 0.f) h = 0.f;
  area[t] = w * h;
  valid[t] = (s > -1e30f) ? 1 : 0;
  __syncthreads();

  for (int it = 0; it < MAX_DET; ++it) {
    red_s[t] = valid[t] ? ssh[t] : NEG_BIG;
    red_i[t] = t;
    __syncthreads();
    for (int off = TOPK / 2; off > 0; off >>= 1) {
      if (t < off) {
        const float so = red_s[t + off]; const int io = red_i[t + off];
        if (so > red_s[t] || (so == red_s[t] && io < red_i[t])) {
          red_s[t] = so; red_i[t] = io;
        }
      }
      __syncthreads();
    }
    const int j = red_i[0];
    const int ok = red_s[0] > -1e30f;
    if (ok) {
      const float jx1 = bx[j][0], jy1 = bx[j][1], jx2 = bx[j][2], jy2 = bx[j][3];
      const float ja = area[j]; const int jc = cls[j];
      const float ltx = jx1 > bx[t][0] ? jx1 : bx[t][0];
      const float lty = jy1 > bx[t][1] ? jy1 : bx[t][1];
      const float rbx = jx2 < bx[t][2] ? jx2 : bx[t][2];
      const float rby = jy2 < bx[t][3] ? jy2 : bx[t][3];
      float iw = rbx - ltx; if (iw < 0.f) iw = 0.f;
      float ih = rby - lty; if (ih < 0.f) ih = 0.f;
      const float inter = iw * ih;
      const float iou = inter / (ja + area[t] - inter + 1e-9f);
      if (iou > IOU_THR && jc == cls[t]) valid[t] = 0;
      if (t == j) valid[t] = 0;
    }
    if (t == 0) {
      float* row = out + ((size_t)(b * MAX_DET + it)) * OUT_ROW;
      if (ok) {
        row[0] = bx[j][0]; row[1] = bx[j][1]; row[2] = bx[j][2]; row[3] = bx[j][3];
        row[4] = ssh[j];   row[5] = (float)cls[j];
        didx_out[b * MAX_DET + it] = anc[j];
      } else {
        row[0] = row[1] = row[2] = row[3] = row[4] = row[5] = 0.f;
        didx_out[b * MAX_DET + it] = 0;
      }
    }
    __syncthreads();
  }
}

// ---------------------------------------------------------------------------
// Kernel 7: masks. Per detection (400 blocks): ROI-align of 5-ch basis bank,
// bilinear 14->160 upsample of attention coeffs, channel softmax, sigmoid.
// Attn coefficients (980 f32) are staged into LDS with CDNA5 async
// global->LDS copies (ASYNCcnt path), overlapping the ROI setup; basis
// planes are prefetched (global_prefetch_b8) since each block re-gathers
// from a 512 KB/batch bank resident in the 192 MB L2.
// ---------------------------------------------------------------------------
__global__ __launch_bounds__(256) void k_masks(const float* __restrict__ boxes,
                                               const float* __restrict__ attn,
                                               const float* __restrict__ bases,
                                               const float* __restrict__ sem,
                                               const int* __restrict__ didx,
                                               float* __restrict__ out) {
  __shared__ float sA[ATTN_LEN];
  const int d = blockIdx.x;
  const int b = d / MAX_DET;
  const int an = didx[d];

  // --- async stage attn -> LDS (tracked by ASYNCcnt) ---
  const float* ap = attn + ((size_t)b * NA + an) * ATTN_LEN;
  for (int i = threadIdx.x; i < ATTN_LEN; i += blockDim.x) {
    const unsigned lds_addr = (unsigned)(size_t)(&sA[i]);   // low 32b = LDS offset
    const unsigned long long gaddr = (unsigned long long)(size_t)(ap + i);
    asm volatile("global_load_async_to_lds_b32 %0, %1, off"
                 :: "v"(lds_addr), "v"(gaddr) : "memory");
  }

  const float* featc[NUM_BASE];
#pragma unroll
  for (int c = 0; c < NUM_BASE - 1; ++c)
    featc[c] = bases + ((size_t)b * (NUM_BASE - 1) + c) * (FH * FW);
  featc[NUM_BASE - 1] = sem + (size_t)b * (FH * FW);
#pragma unroll
  for (int c = 0; c < NUM_BASE; ++c)
    __builtin_prefetch(featc[c] + threadIdx.x * 16, 0, 0);  // global_prefetch_b8

  const float* bp = boxes + ((size_t)b * NA + an) * 4;
  const float rx1 = bp[0] * ROI_SCALE - 0.5f;
  const float ry1 = bp[1] * ROI_SCALE - 0.5f;
  const float rx2 = bp[2] * ROI_SCALE - 0.5f;
  const float ry2 = bp[3] * ROI_SCALE - 0.5f;
  const float stx = (rx2 - rx1) * (1.0f / (float)FW);
  const float sty = (ry2 - ry1) * (1.0f / (float)FH);

  asm volatile("s_wait_asynccnt 0x0" ::: "memory");         // attn landed in LDS
  __syncthreads();

  float* orow = out + (size_t)d * OUT_ROW + 6;
  for (int p = threadIdx.x; p < FH * FW; p += blockDim.x) {
    const int py = p / FW, px = p % FW;
    // --- ROI-align sample coordinate ---
    const float cyf = ry1 + (py + 0.5f) * sty;
    const float cxf = rx1 + (px + 0.5f) * stx;
    const bool vy = (cyf > -1.0f) && (cyf < (float)FH);
    const bool vx = (cxf > -1.0f) && (cxf < (float)FW);
    const float cy = fminf(fmaxf(cyf, 0.0f), (float)(FH - 1));
    const float cx = fminf(fmaxf(cxf, 0.0f), (float)(FW - 1));
    const int y0 = (int)floorf(cy);
    const int x0 = (int)floorf(cx);
    const int y1i = (y0 + 1 > FH - 1) ? FH - 1 : y0 + 1;
    const int x1i = (x0 + 1 > FW - 1) ? FW - 1 : x0 + 1;
    const float wy = cy - (float)y0, wx = cx - (float)x0;
    // --- attention upsample coordinate (half-pixel, edge clamp) ---
    const float ayf = (py + 0.5f) * ((float)ATTN_RES / (float)FH) - 0.5f;
    const float axf = (px + 0.5f) * ((float)ATTN_RES / (float)FW) - 0.5f;
    const float lyf = floorf(ayf), lxf = floorf(axf);
    const float wy2 = ayf - lyf, wx2 = axf - lxf;
    int iy0 = (int)lyf, ix0 = (int)lxf;
    int iy1 = iy0 + 1, ix1 = ix0 + 1;
    iy0 = iy0 < 0 ? 0 : (iy0 > ATTN_RES - 1 ? ATTN_RES - 1 : iy0);
    iy1 = iy1 < 0 ? 0 : (iy1 > ATTN_RES - 1 ? ATTN_RES - 1 : iy1);
    ix0 = ix0 < 0 ? 0 : (ix0 > ATTN_RES - 1 ? ATTN_RES - 1 : ix0);
    ix1 = ix1 < 0 ? 0 : (ix1 > ATTN_RES - 1 ? ATTN_RES - 1 : ix1);

    float coef[NUM_BASE];
    float mx = NEG_BIG;
#pragma unroll
    for (int c = 0; c < NUM_BASE; ++c) {
      const float* am = sA + c * (ATTN_RES * ATTN_RES);
      const float v = am[iy0 * ATTN_RES + ix0] * (1.f - wy2) * (1.f - wx2)
                    + am[iy0 * ATTN_RES + ix1] * (1.f - wy2) * wx2
                    + am[iy1 * ATTN_RES + ix0] * wy2 * (1.f - wx2)
                    + am[iy1 * ATTN_RES + ix1] * wy2 * wx2;
      coef[c] = v;
      mx = fmaxf(mx, v);
    }
    float se = 0.f;
#pragma unroll
    for (int c = 0; c < NUM_BASE; ++c) { coef[c] = expf(coef[c] - mx); se += coef[c]; }
    const float inv = 1.0f / se;

    const bool vld = vy && vx;
    float acc = 0.f;
#pragma unroll
    for (int c = 0; c < NUM_BASE; ++c) {
      float pv = 0.f;
      if (vld) {
        const float* f = featc[c];
        pv = f[y0  * FW + x0 ] * (1.f - wy) * (1.f - wx)
           + f[y0  * FW + x1i] * (1.f - wy) * wx
           + f[y1i * FW + x0 ] * wy * (1.f - wx)
           + f[y1i * FW + x1i] * wy * wx;
      }
      acc += pv * coef[c] * inv;
    }
    orow[p] = 1.0f / (1.0f + expf(-acc));
  }
}

// ---------------------------------------------------------------------------
// Workspace layout (bytes)
// ---------------------------------------------------------------------------
#define OFF_BOXES  ((size_t)0)                         // 4*25200*4 f32 = 1,612,800 B
#define OFF_HIST   ((size_t)1612800)                   // 4*2048 i32   =    32,768 B
#define OFF_THRESH (OFF_HIST + 32768)                  // 4 i32 (pad 256)
#define OFF_COUNT  (OFF_THRESH + 256)                  // 4 i32 (pad 256)
#define OFF_CSC    (OFF_COUNT + 256)                   // 4*4096 f32   =    65,536 B
#define OFF_CIX    (OFF_CSC + 65536)                   // 4*4096 i32   =    65,536 B
#define OFF_TOPS   (OFF_CIX + 65536)                   // 4*1024 f32   =    16,384 B
#define OFF_TOPI   (OFF_TOPS + 16384)                  // 4*1024 i32   =    16,384 B
#define OFF_DIDX   (OFF_TOPI + 16384)                  // 400 i32

extern "C" void kernel_launch(void* const* d_in, const int* in_sizes, int n_in,
                              void* d_out, int out_size, void* d_ws, size_t ws_size,
                              hipStream_t stream) {
  (void)in_sizes; (void)n_in; (void)out_size; (void)ws_size;
  const float* test  = (const float*)d_in[0];
  const float* attn  = (const float*)d_in[1];
  const float* bases = (const float*)d_in[2];
  const float* sem   = (const float*)d_in[3];
  float* out = (float*)d_out;
  char* ws = (char*)d_ws;

  float* boxes  = (float*)(ws + OFF_BOXES);
  int*   hist   = (int*)  (ws + OFF_HIST);
  int*   thresh = (int*)  (ws + OFF_THRESH);
  int*   count  = (int*)  (ws + OFF_COUNT);
  float* cand_s = (float*)(ws + OFF_CSC);
  int*   cand_i = (int*)  (ws + OFF_CIX);
  float* top_s  = (float*)(ws + OFF_TOPS);
  int*   top_i  = (int*)  (ws + OFF_TOPI);
  int*   didx   = (int*)  (ws + OFF_DIDX);

  // zero hist + thresh + count (contiguous) each call -> deterministic
  hipMemsetAsync(ws + OFF_HIST, 0, 32768 + 256 + 256, stream);

  const int tiles = NBATCH * (NA / 16);
  k_boxes_wmma<<<(tiles + 7) / 8, 256, 0, stream>>>(test, boxes);

  dim3 gh((NA + 255) / 256, NBATCH);
  k_hist<<<gh, 256, 0, stream>>>(test, hist);
  k_thresh<<<1, 32, 0, stream>>>(hist, thresh);
  k_collect<<<gh, 256, 0, stream>>>(test, thresh, count, cand_s, cand_i);
  k_sortk<<<NBATCH, 1024, 0, stream>>>(cand_s, cand_i, count, top_s, top_i);
  k_nms<<<NBATCH, 1024, 0, stream>>>(top_s, top_i, boxes, out, didx);
  k_masks<<<NBATCH * MAX_DET, 256, 0, stream>>>(boxes, attn, bases, sem, didx, out);
}